// Embed2GraphByClique_53420803228031
// MI455X (gfx1250) — compile-verified
//
#include <hip/hip_runtime.h>

typedef __attribute__((ext_vector_type(2))) float v2f;
typedef __attribute__((ext_vector_type(8))) float v8f;

#define N_TOK 360
#define D_DIM 128
#define B_SZ  4
#define M_ROWS (B_SZ * N_TOK) /* 1440 */

// ---------------------------------------------------------------------------
// Kernel 1: dual projection GEMM via V_WMMA_F32_16X16X4_F32 (exact fp32 path).
// Grid covers 90 m-tiles x (8 n-tiles x 2 weight halves) = 1440 waves.
// Each wave owns one 16x16 tile; K=128 -> 32 wmma steps.
// Half 0 writes xi + b1 (b1 folded); half 1 writes xj.
// ---------------------------------------------------------------------------
__global__ __launch_bounds__(256) void gemm_wmma(const float* __restrict__ x,
                                                 const float* __restrict__ W1,
                                                 const float* __restrict__ b1,
                                                 float* __restrict__ wsA,
                                                 float* __restrict__ wsC) {
    const int wid  = threadIdx.x >> 5;
    const int lane = threadIdx.x & 31;
    const int gw   = blockIdx.x * 8 + wid;   // 0..1439
    const int mt   = gw >> 4;                // 0..89
    const int nt16 = gw & 15;
    const int half = nt16 >> 3;              // which 128-row half of W1
    const int nt   = nt16 & 7;
    const int mbase = mt * 16;
    const int nbase = nt * 16;
    const int lm    = lane & 15;
    const int koff  = (lane < 16) ? 0 : 2;   // A/B K-pair per ISA VGPR layout

    const float* xrow = x + (mbase + lm) * D_DIM;
    const float* wb   = W1 + half * D_DIM * D_DIM;

    v8f acc = {0.f, 0.f, 0.f, 0.f, 0.f, 0.f, 0.f, 0.f};
#pragma unroll
    for (int k = 0; k < D_DIM; k += 4) {
        // A 16x4 fragment: lane holds (K=k+koff, K=k+koff+1) of row M=lm
        v2f a = *(const v2f*)(xrow + k + koff);
        // B 4x16 fragment: VGPR0 = row k+koff, VGPR1 = row k+koff+1, col N=lm
        v2f b;
        b.x = wb[(k + koff)     * D_DIM + nbase + lm];
        b.y = wb[(k + koff + 1) * D_DIM + nbase + lm];
        acc = __builtin_amdgcn_wmma_f32_16x16x4_f32(
            /*neg_a=*/false, a, /*neg_b=*/false, b,
            /*c_mod=*/(short)0, acc, /*reuse_a=*/false, /*reuse_b=*/false);
    }

    float* dst = half ? wsC : wsA;
    const float badd = half ? 0.0f : b1[nbase + lm];
    const int moff = (lane < 16) ? 0 : 8;
#pragma unroll
    for (int r = 0; r < 8; ++r) {
        dst[(mbase + r + moff) * D_DIM + nbase + lm] = acc[r] + badd;
    }
}

// ---------------------------------------------------------------------------
// Kernel 2: pairwise logits. 16x16 (i,j) tile per 256-thread block.
// A/C tiles staged in LDS (pitch 132 to avoid stride-128 bank conflicts).
// logits[b,i,j] = sum_d relu(A[i,d] + C[j,d]) * W2[d] + b2 ; diagonal -> 0.
// ---------------------------------------------------------------------------
__global__ __launch_bounds__(256) void pair_logits(const float* __restrict__ wsA,
                                                   const float* __restrict__ wsC,
                                                   const float* __restrict__ W2,
                                                   const float* __restrict__ b2,
                                                   float* __restrict__ out) {
    __shared__ float As[16][132];
    __shared__ float Cs[16][132];
    __shared__ float w2s[128];

    const int b  = blockIdx.z;
    const int i0 = blockIdx.x * 16;
    const int j0 = blockIdx.y * 16;
    const int tid = threadIdx.x;

    for (int idx = tid; idx < 16 * 128; idx += 256) {
        const int r = idx >> 7, c = idx & 127;
        const int gi = i0 + r, gj = j0 + r;
        As[r][c] = (gi < N_TOK) ? wsA[(b * N_TOK + gi) * D_DIM + c] : 0.0f;
        Cs[r][c] = (gj < N_TOK) ? wsC[(b * N_TOK + gj) * D_DIM + c] : 0.0f;
    }
    if (tid < 128) w2s[tid] = W2[tid];
    __syncthreads();

    const int tx = tid & 15;   // local j
    const int ty = tid >> 4;   // local i
    float acc = 0.0f;
#pragma unroll 4
    for (int d = 0; d < D_DIM; ++d) {
        float v = As[ty][d] + Cs[tx][d];
        v = fmaxf(v, 0.0f);
        acc = fmaf(v, w2s[d], acc);
    }

    const int i = i0 + ty, j = j0 + tx;
    if (i < N_TOK && j < N_TOK) {
        float lg = acc + b2[0];
        if (i == j) lg = 0.0f;   // logits * (1 - eye): zero logit stays in softmax
        out[(b * N_TOK + i) * N_TOK + j] = lg;
    }
}

// ---------------------------------------------------------------------------
// Kernel 3: numerically-stable row softmax, in place on d_out.
// One 128-thread block per (b,i) row of 360 logits.
// ---------------------------------------------------------------------------
__global__ __launch_bounds__(128) void row_softmax(float* __restrict__ out) {
    __shared__ float red[128];
    float* p = out + (size_t)blockIdx.x * N_TOK;
    const int tid = threadIdx.x;

    float vals[3];
    float m = -3.4e38f;
#pragma unroll
    for (int t = 0; t < 3; ++t) {
        const int j = tid + t * 128;
        vals[t] = (j < N_TOK) ? p[j] : -3.4e38f;
        m = fmaxf(m, vals[t]);
    }
    red[tid] = m;
    __syncthreads();
    for (int s = 64; s > 0; s >>= 1) {
        if (tid < s) red[tid] = fmaxf(red[tid], red[tid + s]);
        __syncthreads();
    }
    m = red[0];
    __syncthreads();

    float sum = 0.0f;
#pragma unroll
    for (int t = 0; t < 3; ++t) {
        const int j = tid + t * 128;
        if (j < N_TOK) {
            vals[t] = expf(vals[t] - m);
            sum += vals[t];
        }
    }
    red[tid] = sum;
    __syncthreads();
    for (int s = 64; s > 0; s >>= 1) {
        if (tid < s) red[tid] += red[tid + s];
        __syncthreads();
    }
    const float inv = 1.0f / red[0];
#pragma unroll
    for (int t = 0; t < 3; ++t) {
        const int j = tid + t * 128;
        if (j < N_TOK) p[j] = vals[t] * inv;
    }
}

// ---------------------------------------------------------------------------
extern "C" void kernel_launch(void* const* d_in, const int* in_sizes, int n_in,
                              void* d_out, int out_size, void* d_ws, size_t ws_size,
                              hipStream_t stream) {
    (void)in_sizes; (void)n_in; (void)out_size; (void)ws_size;
    const float* x  = (const float*)d_in[0];
    const float* W1 = (const float*)d_in[1];
    const float* b1 = (const float*)d_in[2];
    const float* W2 = (const float*)d_in[3];
    const float* b2 = (const float*)d_in[4];
    float* out = (float*)d_out;

    float* wsA = (float*)d_ws;             // 1440 x 128 fp32 : xi + b1
    float* wsC = wsA + M_ROWS * D_DIM;     // 1440 x 128 fp32 : xj

    // 1440 waves total, 8 waves (256 threads) per block -> 180 blocks
    gemm_wmma<<<dim3(M_ROWS * 16 / 16 / 8), 256, 0, stream>>>(x, W1, b1, wsA, wsC);

    dim3 pg((N_TOK + 15) / 16, (N_TOK + 15) / 16, B_SZ);   // (23,23,4)
    pair_logits<<<pg, 256, 0, stream>>>(wsA, wsC, W2, b2, out);

    row_softmax<<<dim3(B_SZ * N_TOK), 128, 0, stream>>>(out);
}